// GNNModel_29454885716720
// MI455X (gfx1250) — compile-verified
//
#include <hip/hip_runtime.h>
#include <hip/hip_bf16.h>

// ---------------------------------------------------------------------------
// 2-layer GCN on MI455X (gfx1250, wave32).
// Memory/atomic-bound scatter-gather; dense per-node GEMMs use
// V_WMMA_F32_16X16X4_F32 (one wave = 16 nodes x 16 features tile).
// Node-state buffers (~30MB) + edge stream (51MB) fit in the 192MB L2, so
// gathers/atomics resolve at L2 bandwidth; HBM floor ~= one edge sweep.
// ---------------------------------------------------------------------------

typedef __attribute__((ext_vector_type(2))) float v2f;
typedef __attribute__((ext_vector_type(4))) float v4f;
typedef __attribute__((ext_vector_type(8))) float v8f;

// ---------------- degree / normalization ----------------

__global__ __launch_bounds__(256) void k_count(const int* __restrict__ dst,
                                               unsigned E, int* __restrict__ cnt) {
    unsigned e = blockIdx.x * blockDim.x + threadIdx.x;
    if (e < E) atomicAdd(&cnt[dst[e]], 1);
}

__global__ __launch_bounds__(256) void k_dinv(const int* __restrict__ cnt,
                                              float* __restrict__ dinv, unsigned n) {
    unsigned i = blockIdx.x * blockDim.x + threadIdx.x;
    if (i < n) {
        float deg = (float)cnt[i] + 1.0f;   // +1 self loop, always > 0
        dinv[i] = rsqrtf(deg);
    }
}

// ---------------- layer-1 GEMM: h1[N,16] = x[N,3(pad 4)] @ W1[3(pad 4),16] ----------------
// Host guarantees n >= 16. Tail tile overlaps (rewrites identical values).
// A layout (16x4 f32): lanes 0-15 -> {K0,K1}, lanes 16-31 -> {K2,K3}, M = lane&15.
// B layout (4x16 f32): VGPR0 row K=(lane<16?0:2), VGPR1 row K=(lane<16?1:3), N = lane&15.
// C/D: VGPR j, lane l -> M = (l<16 ? j : j+8), N = l&15.

__global__ __launch_bounds__(256) void k_gemm1(const float* __restrict__ x,
                                               const float* __restrict__ W1, // [3,16]
                                               float* __restrict__ h1,       // [N,16]
                                               unsigned n) {
    const unsigned lane = threadIdx.x & 31;
    const unsigned wave = (blockIdx.x * blockDim.x + threadIdx.x) >> 5;
    unsigned base = wave * 16u;
    if (base + 16u > n) {                 // wave-uniform: EXEC stays all-ones
        if (base >= n) return;
        base = n - 16u;                   // overlap tail tile (same values)
    }

    const unsigned nn = lane & 15u;
    const unsigned row = base + nn;       // always < n
    const bool lo = (lane < 16);

    // Branchless operand fetch: every lane does an in-bounds load, pad via cndmask.
    const unsigned ka = lo ? 0u : 2u;
    float ax = x[row * 3u + ka];
    float ay = x[row * 3u + (lo ? 1u : 2u)];      // hi lanes: dummy in-bounds load
    v2f a; a.x = ax; a.y = lo ? ay : 0.0f;        // K=3 pad

    float bx = W1[ka * 16u + nn];
    float by = W1[(lo ? 1u : 0u) * 16u + nn];     // hi lanes: dummy in-bounds load
    v2f b; b.x = bx; b.y = lo ? by : 0.0f;        // K=3 pad

    v8f c = {};
    c = __builtin_amdgcn_wmma_f32_16x16x4_f32(false, a, false, b,
                                              (short)0, c, false, false);

    // 8 unconditional coalesced stores, 32-bit offset math.
    const unsigned off = (base + (lo ? 0u : 8u)) * 16u + nn;
    #pragma unroll
    for (unsigned j = 0; j < 8; ++j) h1[off + j * 16u] = c[j];
}

// scalar fallback for n < 16 (never taken for the real workload)
__global__ __launch_bounds__(256) void k_gemm1_small(const float* __restrict__ x,
                                                     const float* __restrict__ W1,
                                                     float* __restrict__ h1,
                                                     unsigned n) {
    unsigned i = blockIdx.x * blockDim.x + threadIdx.x;
    if (i >= n * 16u) return;
    unsigned row = i >> 4, f = i & 15u;
    float acc = 0.0f;
    #pragma unroll
    for (unsigned k = 0; k < 3; ++k) acc += x[row * 3u + k] * W1[k * 16u + f];
    h1[i] = acc;
}

// ---------------- layer-1 propagate: agg[d] += h1[s] * dinv[s]*dinv[d] ----------------
// 16 lanes per edge, one f32 atomic per (edge, feature). 32-bit index math only.

__global__ __launch_bounds__(256) void k_mp16(const int* __restrict__ src,
                                              const int* __restrict__ dst,
                                              const float* __restrict__ dinv,
                                              const float* __restrict__ h1,
                                              float* __restrict__ agg,
                                              unsigned E) {
    unsigned gid = blockIdx.x * blockDim.x + threadIdx.x;   // E*16 = 102.4M < 2^31
    if (gid >= E * 16u) return;
    unsigned e = gid >> 4, f = gid & 15u;
    unsigned s = (unsigned)src[e], d = (unsigned)dst[e];
    float w = dinv[s] * dinv[d];
    atomicAdd(&agg[d * 16u + f], h1[s * 16u + f] * w);
}

// ---------------- self-loop + bias + ReLU (in place on agg, float4) ----------------

__global__ __launch_bounds__(256) void k_relu_bias(float* __restrict__ agg,
                                                   const float* __restrict__ h1,
                                                   const float* __restrict__ dinv,
                                                   const float* __restrict__ b1,
                                                   unsigned n) {
    unsigned gid = blockIdx.x * blockDim.x + threadIdx.x;   // one float4 per thread
    if (gid >= n * 4u) return;
    unsigned i = gid >> 2, q = (gid & 3u) * 4u;
    float di = dinv[i];
    float d2 = di * di;
    v4f va = *(const v4f*)(agg + i * 16u + q);
    v4f vh = *(const v4f*)(h1  + i * 16u + q);
    v4f vb = *(const v4f*)(b1 + q);
    v4f v;
    #pragma unroll
    for (int t = 0; t < 4; ++t) {
        float u = va[t] + vh[t] * d2 + vb[t];
        v[t] = u > 0.0f ? u : 0.0f;
    }
    *(v4f*)(agg + i * 16u + q) = v;
}

// ---------------- layer-2 GEMM: h2[N,3] = hrelu[N,16] @ W2[16,3(pad 16)] ----------------
// Host guarantees n >= 16. Four chained K=4 WMMAs accumulate the K=16 contraction.

__global__ __launch_bounds__(256) void k_gemm2(const float* __restrict__ h,   // [N,16]
                                               const float* __restrict__ W2,  // [16,3]
                                               float* __restrict__ h2,        // [N,3]
                                               unsigned n) {
    const unsigned lane = threadIdx.x & 31;
    const unsigned wave = (blockIdx.x * blockDim.x + threadIdx.x) >> 5;
    unsigned base = wave * 16u;
    if (base + 16u > n) {
        if (base >= n) return;
        base = n - 16u;
    }

    const unsigned nn = lane & 15u;
    const unsigned row = base + nn;
    const bool lo = (lane < 16);
    const bool colok = (nn < 3u);
    const unsigned nc = colok ? nn : 2u;          // clamped in-bounds column

    v8f c = {};
    #pragma unroll
    for (unsigned kc = 0; kc < 4; ++kc) {
        const unsigned k0 = kc * 4u + (lo ? 0u : 2u);
        v2f a; a.x = h[row * 16u + k0];
               a.y = h[row * 16u + k0 + 1];
        float b0 = W2[(k0    ) * 3u + nc];        // unconditional in-bounds loads
        float b1v = W2[(k0 + 1) * 3u + nc];
        v2f b; b.x = colok ? b0 : 0.0f;           // column pad via cndmask
               b.y = colok ? b1v : 0.0f;
        c = __builtin_amdgcn_wmma_f32_16x16x4_f32(false, a, false, b,
                                                  (short)0, c, false, false);
    }

    const unsigned off = (base + (lo ? 0u : 8u)) * 3u + nn;
    if (colok) {                                  // single EXEC-mask region
        #pragma unroll
        for (unsigned j = 0; j < 8; ++j) h2[off + j * 3u] = c[j];
    }
}

// scalar fallback for n < 16 (never taken for the real workload)
__global__ __launch_bounds__(256) void k_gemm2_small(const float* __restrict__ h,
                                                     const float* __restrict__ W2,
                                                     float* __restrict__ h2,
                                                     unsigned n) {
    unsigned i = blockIdx.x * blockDim.x + threadIdx.x;
    if (i >= n * 3u) return;
    unsigned row = i / 3u, f = i - row * 3u;
    float acc = 0.0f;
    #pragma unroll
    for (unsigned k = 0; k < 16; ++k) acc += h[row * 16u + k] * W2[k * 3u + f];
    h2[i] = acc;
}

// ---------------- layer-2 propagate: 3 atomics per edge ----------------

__global__ __launch_bounds__(256) void k_mp3(const int* __restrict__ src,
                                             const int* __restrict__ dst,
                                             const float* __restrict__ dinv,
                                             const float* __restrict__ h2,
                                             float* __restrict__ oagg,
                                             unsigned E) {
    unsigned e = blockIdx.x * blockDim.x + threadIdx.x;
    if (e >= E) return;
    unsigned s = (unsigned)src[e], d = (unsigned)dst[e];
    float w = dinv[s] * dinv[d];
    float m0 = h2[s * 3u + 0] * w;
    float m1 = h2[s * 3u + 1] * w;
    float m2 = h2[s * 3u + 2] * w;
    atomicAdd(&oagg[d * 3u + 0], m0);
    atomicAdd(&oagg[d * 3u + 1], m1);
    atomicAdd(&oagg[d * 3u + 2], m2);
}

// ---------------- finalize: self-loop + bias -> d_out ----------------

__global__ __launch_bounds__(256) void k_final(const float* __restrict__ oagg,
                                               const float* __restrict__ h2,
                                               const float* __restrict__ dinv,
                                               const float* __restrict__ b2,
                                               float* __restrict__ out,
                                               unsigned n) {
    unsigned gid = blockIdx.x * blockDim.x + threadIdx.x;
    if (gid >= n * 3u) return;
    unsigned i = gid / 3u, f = gid - i * 3u;
    float di = dinv[i];
    out[gid] = oagg[gid] + h2[gid] * (di * di) + b2[f];
}

// ---------------------------------------------------------------------------

extern "C" void kernel_launch(void* const* d_in, const int* in_sizes, int n_in,
                              void* d_out, int out_size, void* d_ws, size_t ws_size,
                              hipStream_t stream) {
    const float* x  = (const float*)d_in[0];
    const int*   ei = (const int*)  d_in[1];
    const float* W1 = (const float*)d_in[2];
    const float* b1 = (const float*)d_in[3];
    const float* W2 = (const float*)d_in[4];
    const float* b2 = (const float*)d_in[5];
    float* out = (float*)d_out;

    const unsigned n = (unsigned)(in_sizes[0] / 3);
    const unsigned E = (unsigned)(in_sizes[1] / 2);
    const int* src = ei;
    const int* dst = ei + E;

    // workspace carve-up (~32 MB total for N=200k)
    char* ws = (char*)d_ws;
    int*   cnt  = (int*)  ws; ws += (size_t)n * sizeof(int);
    float* dinv = (float*)ws; ws += (size_t)n * sizeof(float);
    float* h1   = (float*)ws; ws += (size_t)n * 16 * sizeof(float);
    float* agg  = (float*)ws; ws += (size_t)n * 16 * sizeof(float);
    float* h2   = (float*)ws; ws += (size_t)n * 3 * sizeof(float);
    float* oagg = (float*)ws; ws += (size_t)n * 3 * sizeof(float);

    hipMemsetAsync(cnt,  0, (size_t)n * sizeof(int), stream);
    hipMemsetAsync(agg,  0, (size_t)n * 16 * sizeof(float), stream);
    hipMemsetAsync(oagg, 0, (size_t)n * 3 * sizeof(float), stream);

    const unsigned B = 256;

    // degree + symmetric normalization
    k_count<<<(E + B - 1) / B, B, 0, stream>>>(dst, E, cnt);
    k_dinv <<<(n + B - 1) / B, B, 0, stream>>>(cnt, dinv, n);

    // layer 1
    if (n >= 16) {
        unsigned tiles  = (n + 15) / 16;       // one wave per 16-node tile
        unsigned blocks = (tiles + 7) / 8;     // 8 waves / 256-thread block
        k_gemm1<<<blocks, B, 0, stream>>>(x, W1, h1, n);
    } else {
        k_gemm1_small<<<(n * 16 + B - 1) / B, B, 0, stream>>>(x, W1, h1, n);
    }
    {
        unsigned work = E * 16u;               // 102.4M threads
        k_mp16<<<(work + B - 1) / B, B, 0, stream>>>(src, dst, dinv, h1, agg, E);
    }
    {
        unsigned work = n * 4u;                // one float4 per thread
        k_relu_bias<<<(work + B - 1) / B, B, 0, stream>>>(agg, h1, dinv, b1, n);
    }

    // layer 2
    if (n >= 16) {
        unsigned tiles  = (n + 15) / 16;
        unsigned blocks = (tiles + 7) / 8;
        k_gemm2<<<blocks, B, 0, stream>>>(agg, W2, h2, n);
    } else {
        k_gemm2_small<<<(n * 3 + B - 1) / B, B, 0, stream>>>(agg, W2, h2, n);
    }
    k_mp3<<<(E + B - 1) / B, B, 0, stream>>>(src, dst, dinv, h2, oagg, E);
    {
        unsigned work = n * 3u;
        k_final<<<(work + B - 1) / B, B, 0, stream>>>(oagg, h2, dinv, b2, out, n);
    }
}